// LongformerSelfAttention_68101001445921
// MI455X (gfx1250) — compile-verified
//
#include <hip/hip_runtime.h>
#include <math.h>

// Problem constants (from reference)
#define NSEQ   2048
#define DMODEL 1024
#define NH     16
#define HD     64
#define KVD    256     // D_OUT/G columns of k/v projections
#define CCH    17      // chunk count
#define WIN    128
#define KB     32      // GEMM K-block

typedef __attribute__((ext_vector_type(2))) float v2f;
typedef __attribute__((ext_vector_type(8))) float v8f;

// D(16x16,f32) = A(16x4,f32) x B(4x16,f32) + C   -- CDNA5 V_WMMA_F32_16X16X4_F32
static __device__ __forceinline__ v8f wmma_f32(v2f a, v2f b, v8f c) {
  return __builtin_amdgcn_wmma_f32_16x16x4_f32(false, a, false, b, (short)0, c,
                                               false, false);
}

// ---------------------------------------------------------------------------
// Async global -> LDS copy (CDNA5): 16 bytes per lane, tracked by ASYNCcnt.
// The low 32 bits of a flat pointer to __shared__ are the LDS byte offset
// (aperture truncation per ISA 10.2), which is exactly what VDST needs.
// ---------------------------------------------------------------------------
#define USE_ASYNC_LDS_COPY 1

static __device__ __forceinline__ void copy_b128(void* lds, const void* g) {
#if USE_ASYNC_LDS_COPY
  const unsigned l = (unsigned)(uintptr_t)lds;
  asm volatile("global_load_async_to_lds_b128 %0, %1, off"
               :: "v"(l), "v"((unsigned long long)(uintptr_t)g)
               : "memory");
#else
  *(float4*)lds = *(const float4*)g;
#endif
}
static __device__ __forceinline__ void copy_join() {
#if USE_ASYNC_LDS_COPY
  asm volatile("s_wait_asynccnt 0x0" ::: "memory");
#endif
}

// ---------------------------------------------------------------------------
// GEMM: C[M x N] = X[M x K] @ W[K x N] (+ bias).  M mult of 128, N of 64.
// 256 threads = 8 waves; block tile 128(M) x 64(N); wave tile 16 x 64.
// Double-buffered async staging of X and W K-blocks through LDS.
//   Xs stride 36 floats: A-frag b64 reads conflict-free (banks 4*(9i%16))
//   Ws stride 80 floats: B-frag rows k,k+1; lanes 16-31 read rows k+2,k+3 (+160 floats = +32 banks of 64)
// ---------------------------------------------------------------------------
__global__ void gemm_f32_wmma(const float* __restrict__ X,
                              const float* __restrict__ W,
                              const float* __restrict__ bias,
                              float* __restrict__ C, int K, int N) {
  const int tid  = threadIdx.x;
  const int lane = tid & 31;
  const int wave = tid >> 5;
  const int half = lane >> 4;
  const int l16  = lane & 15;
  const int m0   = blockIdx.x * 128;
  const int n0   = blockIdx.y * 64;

  __shared__ float Xs[2][128][36];   // 128 M-rows x KB floats (+pad)
  __shared__ float Ws[2][KB][80];    // KB K-rows x 64 floats (+pad)

  auto stage = [&](int buf, int kb) {
    const int k0 = kb * KB;
    // X tile: 128 rows x 128B  -> 1024 chunks of 16B, 4 per thread
#pragma unroll
    for (int rep = 0; rep < 4; ++rep) {
      const int flat = rep * 256 + tid;
      const int row  = flat >> 3;
      const int ch   = flat & 7;
      copy_b128(&Xs[buf][row][ch * 4],
                X + (size_t)(m0 + row) * K + k0 + ch * 4);
    }
    // W tile: KB rows x 256B -> 512 chunks of 16B, 2 per thread
#pragma unroll
    for (int rep = 0; rep < 2; ++rep) {
      const int flat = rep * 256 + tid;
      const int row  = flat >> 4;
      const int ch   = flat & 15;
      copy_b128(&Ws[buf][row][ch * 4],
                W + (size_t)(k0 + row) * N + n0 + ch * 4);
    }
  };

  v8f acc[4] = {};
  stage(0, 0);
  copy_join();
  __syncthreads();

  const int nkb = K / KB;
  for (int kb = 0; kb < nkb; ++kb) {
    const int cur = kb & 1;
    if (kb + 1 < nkb) stage(1 - cur, kb + 1);   // overlap DMA with compute
#pragma unroll
    for (int kk = 0; kk < KB; kk += 4) {
      v2f a;
      a.x = Xs[cur][wave * 16 + l16][kk + 2 * half];
      a.y = Xs[cur][wave * 16 + l16][kk + 2 * half + 1];
#pragma unroll
      for (int n = 0; n < 4; ++n) {
        v2f b;
        b.x = Ws[cur][kk + 2 * half][16 * n + l16];
        b.y = Ws[cur][kk + 2 * half + 1][16 * n + l16];
        acc[n] = wmma_f32(a, b, acc[n]);
      }
    }
    copy_join();       // next buffer's DMA done (this wave's share)
    __syncthreads();   // everyone done reading cur + staging next
  }

#pragma unroll
  for (int n = 0; n < 4; ++n) {
    const int col = n0 + 16 * n + l16;
    const float bb = bias ? bias[col] : 0.0f;
#pragma unroll
    for (int r = 0; r < 8; ++r) {
      const int row = m0 + wave * 16 + r + 8 * half;
      C[(size_t)row * N + col] = acc[n][r] + bb;
    }
  }
}

// ---------------------------------------------------------------------------
// Global attention: causal + ALiBi, GQA kv-group = h%4 (tile() semantics).
// grid (H=16, N/128); 8 waves each own 16 query rows.  K/V 16-row tiles are
// staged once per block (shared by all waves) via double-buffered async DMA.
// Waves past their causal limit co-stage + barrier but skip compute.
// Writes only rows with gmask==1.
// ---------------------------------------------------------------------------
__global__ void global_attn(const float* __restrict__ qg,
                            const float* __restrict__ kg,
                            const float* __restrict__ vg,
                            const int* __restrict__ gmask,
                            float* __restrict__ ctx) {
  const int h    = blockIdx.x;
  const int tid  = threadIdx.x;
  const int lane = tid & 31;
  const int wave = tid >> 5;
  const int half = lane >> 4;
  const int l16  = lane & 15;
  const int qb    = blockIdx.y * 128;
  const int qbase = qb + wave * 16;
  const int jlim  = qbase >> 4;         // last causal key tile for this wave
  const int jtmax = (qb >> 4) + 8;      // tiles staged by the block
  const float slope = exp2f(-0.5f * (float)(h + 1));
  const int kvc = (h & 3) * 64;

  __shared__ float Kt[2][16][68];
  __shared__ float Vt[2][16][80];
  __shared__ float Pl[8][16][17];       // per-wave P tile (D->A relayout)

  auto stageKV = [&](int buf, int jt) {
    const int j0  = jt * 16;
    const int row = tid >> 4;           // 0..15
    const int ch  = tid & 15;           // 16B chunk in 256B row
    copy_b128(&Kt[buf][row][ch * 4], kg + (size_t)(j0 + row) * KVD + kvc + ch * 4);
    copy_b128(&Vt[buf][row][ch * 4], vg + (size_t)(j0 + row) * KVD + kvc + ch * 4);
  };

  // Preload the 16 A-fragments of Q (K-dim = HD = 64 -> 16 k-steps of 4)
  float qa0[16], qa1[16];
  {
    const float* qrow = qg + (size_t)(qbase + l16) * DMODEL + h * HD + 2 * half;
#pragma unroll
    for (int s = 0; s < 16; ++s) { qa0[s] = qrow[4 * s]; qa1[s] = qrow[4 * s + 1]; }
  }

  v8f O[4] = {};
  float mrow[8], lrow[8];
#pragma unroll
  for (int r = 0; r < 8; ++r) { mrow[r] = -3.0e38f; lrow[r] = 0.0f; }

  stageKV(0, 0);
  copy_join();
  __syncthreads();

  for (int jt = 0; jt < jtmax; ++jt) {
    const int cur = jt & 1;
    if (jt + 1 < jtmax) stageKV(1 - cur, jt + 1);
    if (jt <= jlim) {
      const int j0 = jt * 16;
      // S = Q @ K^T : B[d][j] = K[j][d], read from LDS tile
      v8f S = {};
#pragma unroll
      for (int s = 0; s < 16; ++s) {
        v2f a; a.x = qa0[s]; a.y = qa1[s];
        v2f b;
        b.x = Kt[cur][l16][4 * s + 2 * half];
        b.y = Kt[cur][l16][4 * s + 2 * half + 1];
        S = wmma_f32(a, b, S);
      }
      const int j = j0 + l16;
#pragma unroll
      for (int r = 0; r < 8; ++r) {
        const int i = qbase + r + 8 * half;
        float sc = (j > i) ? -1.0e30f : (S[r] + slope * (float)(j - i)) * 0.125f;
        float mx = sc;
#pragma unroll
        for (int off = 1; off < 16; off <<= 1) mx = fmaxf(mx, __shfl_xor(mx, off, 32));
        const float mnew = fmaxf(mrow[r], mx);
        const float corr = __expf(mrow[r] - mnew);
        const float p    = __expf(sc - mnew);
        float sum = p;
#pragma unroll
        for (int off = 1; off < 16; off <<= 1) sum += __shfl_xor(sum, off, 32);
        lrow[r] = lrow[r] * corr + sum;
        mrow[r] = mnew;
#pragma unroll
        for (int n = 0; n < 4; ++n) O[n][r] *= corr;
        Pl[wave][r + 8 * half][l16] = p;  // same-wave LDS: in-order, no barrier
      }
      // O += P @ V  (A = P via LDS, B = V tile from LDS)
#pragma unroll
      for (int kk = 0; kk < 16; kk += 4) {
        v2f a;
        a.x = Pl[wave][l16][kk + 2 * half];
        a.y = Pl[wave][l16][kk + 2 * half + 1];
#pragma unroll
        for (int n = 0; n < 4; ++n) {
          v2f b;
          b.x = Vt[cur][kk + 2 * half][16 * n + l16];
          b.y = Vt[cur][kk + 2 * half + 1][16 * n + l16];
          O[n] = wmma_f32(a, b, O[n]);
        }
      }
    }
    copy_join();
    __syncthreads();
  }

#pragma unroll
  for (int r = 0; r < 8; ++r) {
    const int i = qbase + r + 8 * half;
    if (gmask[i] == 1) {
      const float inv = 1.0f / lrow[r];
#pragma unroll
      for (int n = 0; n < 4; ++n)
        ctx[(size_t)i * DMODEL + h * HD + 16 * n + l16] = O[n][r] * inv;
    }
  }
}

// ---------------------------------------------------------------------------
// Local (windowed) attention over the scrambled reshape.  Chunk (h,c):
//   source row n(w) = 8*(h*17+c) + w/16  (constant per 16-row tile)
//   keys   K[w][d] = ql[n(w)][ (w%16)*64 + d ]
//   query  Q[w][d] = kl[n(w)][ ((w%16)&3)*64 + d ]   (GQA tile -> mod 256)
//   value  V[w][d] = vl[n(w)][ ((w%16)&3)*64 + d ]
// Blocks with h*17+c >= 256 are entirely in the zero pad -> ctxA = 0.
// ---------------------------------------------------------------------------
__global__ void local_attn(const float* __restrict__ ql,
                           const float* __restrict__ kl,
                           const float* __restrict__ vl,
                           float* __restrict__ ctxA) {
  const int h  = blockIdx.x;
  const int c  = blockIdx.y;
  const int t1 = h * CCH + c;
  const int lane = threadIdx.x & 31;
  const int wave = threadIdx.x >> 5;
  const int half = lane >> 4;
  const int l16  = lane & 15;
  float* outBase = ctxA + (size_t)t1 * WIN * HD;

  if (t1 >= 256) {  // fully padded chunk (h==15, c>=1): context exactly zero
#pragma unroll
    for (int n = 0; n < 4; ++n)
#pragma unroll
      for (int r = 0; r < 8; ++r) {
        const int i = wave * 16 + r + 8 * half;
        outBase[(size_t)i * HD + 16 * n + l16] = 0.0f;
      }
    return;
  }

  const float slope = exp2f(-0.5f * (float)(h + 1));
  float qa0[16], qa1[16];
  {
    const float* qrow = kl + (size_t)(8 * t1 + wave) * KVD + (l16 & 3) * 64 + 2 * half;
#pragma unroll
    for (int s = 0; s < 16; ++s) { qa0[s] = qrow[4 * s]; qa1[s] = qrow[4 * s + 1]; }
  }

  v8f O[4] = {};
  float mrow[8], lrow[8];
#pragma unroll
  for (int r = 0; r < 8; ++r) { mrow[r] = -3.0e38f; lrow[r] = 0.0f; }

  __shared__ float Pl[8][16][17];

  const int jtiles = wave + 1;  // causal within the 128-window
  for (int jt = 0; jt < jtiles; ++jt) {
    const int j0 = jt * 16;
    const int nk = 8 * t1 + jt;
    v8f S = {};
    const float* krow = ql + (size_t)nk * DMODEL + l16 * 64 + 2 * half;
#pragma unroll
    for (int s = 0; s < 16; ++s) {
      v2f a; a.x = qa0[s]; a.y = qa1[s];
      v2f b; b.x = krow[4 * s]; b.y = krow[4 * s + 1];
      S = wmma_f32(a, b, S);
    }
    const int j = j0 + l16;
#pragma unroll
    for (int r = 0; r < 8; ++r) {
      const int i = wave * 16 + r + 8 * half;
      float sc = (j > i) ? -1.0e30f : (S[r] + slope * (float)(j - i)) * 0.125f;
      float mx = sc;
#pragma unroll
      for (int off = 1; off < 16; off <<= 1) mx = fmaxf(mx, __shfl_xor(mx, off, 32));
      const float mnew = fmaxf(mrow[r], mx);
      const float corr = __expf(mrow[r] - mnew);
      const float p    = __expf(sc - mnew);
      float sum = p;
#pragma unroll
      for (int off = 1; off < 16; off <<= 1) sum += __shfl_xor(sum, off, 32);
      lrow[r] = lrow[r] * corr + sum;
      mrow[r] = mnew;
#pragma unroll
      for (int n = 0; n < 4; ++n) O[n][r] *= corr;
      Pl[wave][r + 8 * half][l16] = p;
    }
    // V rows repeat with period 4 inside the tile -> one B-frag per n
    v2f bv[4];
    {
      const float* vr = vl + (size_t)nk * KVD + (2 * half) * 64 + l16;
#pragma unroll
      for (int n = 0; n < 4; ++n) { bv[n].x = vr[16 * n]; bv[n].y = vr[16 * n + 64]; }
    }
#pragma unroll
    for (int kk = 0; kk < 16; kk += 4) {
      v2f a;
      a.x = Pl[wave][l16][kk + 2 * half];
      a.y = Pl[wave][l16][kk + 2 * half + 1];
#pragma unroll
      for (int n = 0; n < 4; ++n) O[n] = wmma_f32(a, bv[n], O[n]);
    }
  }
#pragma unroll
  for (int r = 0; r < 8; ++r) {
    const int i = wave * 16 + r + 8 * half;
    const float inv = 1.0f / lrow[r];
#pragma unroll
    for (int n = 0; n < 4; ++n)
      outBase[(size_t)i * HD + 16 * n + l16] = O[n][r] * inv;
  }
}

// lrows[c2][h*64+d] = (1/128) * sum_{t=c2*128..+127} ctxA[h][t%17][t/17][d]
__global__ void local_reduce(const float* __restrict__ ctxA, float* __restrict__ lrows) {
  const int c2 = blockIdx.x;                       // 0..16
  const int f  = blockIdx.y * 256 + threadIdx.x;   // 0..1023
  const int h  = f >> 6, d = f & 63;
  float s = 0.0f;
  const int tbase = c2 * 128;
  for (int u = 0; u < 128; ++u) {
    const int t = tbase + u;
    const int c = t % CCH, w = t / CCH;
    s += ctxA[((size_t)(h * CCH + c) * WIN + w) * HD + d];
  }
  lrows[c2 * DMODEL + f] = s * (1.0f / 128.0f);
}

// ctx[n][f] = gmask ? (filled later by global_attn) : (n<17 ? lrows[n][f] : 0)
__global__ void prefill_ctx(const float* __restrict__ lrows,
                            const int* __restrict__ gmask,
                            float* __restrict__ ctx) {
  const int idx = blockIdx.x * 256 + threadIdx.x;
  const int nrow = idx >> 10, f = idx & 1023;
  float v = 0.0f;
  if (gmask[nrow] != 1 && nrow < CCH) v = lrows[nrow * DMODEL + f];
  ctx[idx] = v;
}

// ---------------------------------------------------------------------------
extern "C" void kernel_launch(void* const* d_in, const int* in_sizes, int n_in,
                              void* d_out, int out_size, void* d_ws, size_t ws_size,
                              hipStream_t stream) {
  (void)in_sizes; (void)n_in; (void)out_size; (void)ws_size;
  const float* x     = (const float*)d_in[0];
  const int*   gmask = (const int*)d_in[1];
  const float* W_lq  = (const float*)d_in[2];
  const float* W_lk  = (const float*)d_in[3];
  const float* W_lv  = (const float*)d_in[4];
  const float* W_q   = (const float*)d_in[5];
  const float* W_k   = (const float*)d_in[6];
  const float* W_v   = (const float*)d_in[7];
  const float* W_p   = (const float*)d_in[8];
  const float* b_p   = (const float*)d_in[9];
  float* out = (float*)d_out;

  float* ws   = (float*)d_ws;                  // ~33 MB total
  float* qg   = ws;                            // 2048x1024
  float* kg   = qg + 2097152;                  // 2048x256
  float* vg   = kg + 524288;                   // 2048x256
  float* ql   = vg + 524288;                   // 2048x1024
  float* kl   = ql + 2097152;                  // 2048x256
  float* vl   = kl + 524288;                   // 2048x256
  float* ctxA = vl + 524288;                   // 16*17*128*64
  float* lrows = ctxA + 2228224;               // 17x1024
  float* ctx  = ql;  // alias: ql is dead once local_attn has run

  const dim3 blk(256);
  // Projections (f32 WMMA GEMMs, async-LDS staged)
  gemm_f32_wmma<<<dim3(16, 16), blk, 0, stream>>>(x, W_q,  nullptr, qg, DMODEL, DMODEL);
  gemm_f32_wmma<<<dim3(16,  4), blk, 0, stream>>>(x, W_k,  nullptr, kg, DMODEL, KVD);
  gemm_f32_wmma<<<dim3(16,  4), blk, 0, stream>>>(x, W_v,  nullptr, vg, DMODEL, KVD);
  gemm_f32_wmma<<<dim3(16, 16), blk, 0, stream>>>(x, W_lq, nullptr, ql, DMODEL, DMODEL);
  gemm_f32_wmma<<<dim3(16,  4), blk, 0, stream>>>(x, W_lk, nullptr, kl, DMODEL, KVD);
  gemm_f32_wmma<<<dim3(16,  4), blk, 0, stream>>>(x, W_lv, nullptr, vl, DMODEL, KVD);
  // Local (windowed) attention over scrambled views + reduction to 17 rows
  local_attn<<<dim3(NH, CCH), blk, 0, stream>>>(ql, kl, vl, ctxA);
  local_reduce<<<dim3(CCH, 4), blk, 0, stream>>>(ctxA, lrows);
  // Fill ctx with local/zero rows, then overwrite gmask==1 rows with global attn
  prefill_ctx<<<dim3((NSEQ * DMODEL) / 256), blk, 0, stream>>>(lrows, gmask, ctx);
  global_attn<<<dim3(NH, NSEQ / 128), blk, 0, stream>>>(qg, kg, vg, gmask, ctx);
  // Output projection with bias
  gemm_f32_wmma<<<dim3(16, 16), blk, 0, stream>>>(ctx, W_p, b_p, out, DMODEL, DMODEL);
}